// mixture_model_33543694581882
// MI455X (gfx1250) — compile-verified
//
#include <hip/hip_runtime.h>
#include <math.h>

typedef float v2f __attribute__((ext_vector_type(2)));
typedef float v8f __attribute__((ext_vector_type(8)));

#define D_DIM 256
#define K_DIM 64
#define N_ROWS 4096
#define ROWS_PER_BLOCK 32
#define ROWS_PER_WAVE 16
#define LN2 0.69314718055994530942f

// workspace layout (in floats)
#define WS_MUT   0        // muT  [K][D]  (transposed for per-lane b128 loads)
#define WS_IBT   16384    // invbT[K][D]
#define WS_W     32768    // W    [D][K]  = |mu|*exp(-b) + b
#define WS_CK    49152    // ck   [K]     = pi - colsum(b)
#define WS_PART  49216    // partials [256]

// ---------------------------------------------------------------------------
// Precompute muT, invbT, W
__global__ __launch_bounds__(256) void lmm_prep(const float* __restrict__ mu,
                                                const float* __restrict__ b,
                                                float* __restrict__ ws) {
  int idx = blockIdx.x * 256 + threadIdx.x;   // over D*K, [d][k] order
  int d = idx >> 6, k = idx & 63;
  float bv  = b[idx];
  float muv = mu[idx];
  float ib  = expf(-bv);
  ws[WS_MUT + k * D_DIM + d] = muv;
  ws[WS_IBT + k * D_DIM + d] = ib;
  ws[WS_W + idx] = fabsf(muv) * ib + bv;
}

// ck[k] = pi[k] - sum_d b[d][k]
__global__ __launch_bounds__(64) void lmm_ck(const float* __restrict__ b,
                                             const float* __restrict__ pi,
                                             float* __restrict__ ws) {
  int k = threadIdx.x;
  float s = 0.f;
  for (int d = 0; d < D_DIM; ++d) s += b[d * K_DIM + k];
  ws[WS_CK + k] = pi[k] - s;
}

// ---------------------------------------------------------------------------
// Main: per-wave 16-row tile; dense |x-mu|*invb VALU loop + WMMA mask-correction
__global__ __launch_bounds__(64) void lmm_main(const float* __restrict__ X,
                                               const float* __restrict__ ws,
                                               float* __restrict__ partials) {
  __shared__ float    xc[ROWS_PER_BLOCK * D_DIM];     // NaN-cleaned X tile (32KB)
  __shared__ unsigned mbits[ROWS_PER_BLOCK * 8];      // missing-entry bitmask
  __shared__ int      miss[ROWS_PER_BLOCK];           // missing count per row
  __shared__ float    bsum[2][ROWS_PER_WAVE * K_DIM]; // WMMA tile relayout (8KB)

  const int tid = threadIdx.x;
  if (tid < ROWS_PER_BLOCK) miss[tid] = 0;
  __syncthreads();

  // ---- stage X tile: clean NaN -> 0, build missing bitmask + counts ----
  const int row0g = blockIdx.x * ROWS_PER_BLOCK;
  for (int c = 0; c < 4; ++c) {
    int chunk = tid + 64 * c;            // 0..255 = 32 rows x 8 chunks
    int row = chunk >> 3;
    int ch  = chunk & 7;
    const float* src = X + (size_t)(row0g + row) * D_DIM + ch * 32;
    float* dst = xc + row * D_DIM + ch * 32;
    unsigned mb = 0;
    #pragma unroll 8
    for (int e = 0; e < 32; ++e) {
      float v = src[e];
      bool m = (v != v);                 // isnan
      if (m) mb |= (1u << e);
      dst[e] = m ? 0.0f : v;
    }
    mbits[row * 8 + ch] = mb;
    atomicAdd(&miss[row], (int)__popc(mb));
  }
  __syncthreads();

  const int lane = tid & 31;
  const int wv   = __builtin_amdgcn_readfirstlane(tid >> 5);
  const int rl0  = wv * ROWS_PER_WAVE;   // local row base of this wave
  const int la   = lane & 15;
  const int hi   = lane >> 4;
  const int kl0  = lane, kl1 = lane + 32;

  const float* muT = ws + WS_MUT;
  const float* ibT = ws + WS_IBT;
  const float* Wm  = ws + WS_W;

  float accA[ROWS_PER_WAVE][2];
  #pragma unroll
  for (int r = 0; r < ROWS_PER_WAVE; ++r) { accA[r][0] = 0.f; accA[r][1] = 0.f; }

  v8f cacc[4];
  #pragma unroll
  for (int t = 0; t < 4; ++t)
    cacc[t] = v8f{0.f, 0.f, 0.f, 0.f, 0.f, 0.f, 0.f, 0.f};

  for (int d = 0; d < D_DIM; d += 4) {
    // per-lane mu / invb for k=lane and k=lane+32 (b128 from transposed layout)
    float4 m0 = *(const float4*)(muT + kl0 * D_DIM + d);
    float4 m1 = *(const float4*)(muT + kl1 * D_DIM + d);
    float4 i0 = *(const float4*)(ibT + kl0 * D_DIM + d);
    float4 i1 = *(const float4*)(ibT + kl1 * D_DIM + d);

    // ---- dense abs part: acc += |xc - mu| * invb (mask-free) ----
    #pragma unroll
    for (int r = 0; r < ROWS_PER_WAVE; ++r) {
      float4 xv = *(const float4*)(xc + (rl0 + r) * D_DIM + d); // broadcast ds_load_b128
      float a0 = accA[r][0], a1 = accA[r][1];
      a0 = fmaf(fabsf(xv.x - m0.x), i0.x, a0);
      a0 = fmaf(fabsf(xv.y - m0.y), i0.y, a0);
      a0 = fmaf(fabsf(xv.z - m0.z), i0.z, a0);
      a0 = fmaf(fabsf(xv.w - m0.w), i0.w, a0);
      a1 = fmaf(fabsf(xv.x - m1.x), i1.x, a1);
      a1 = fmaf(fabsf(xv.y - m1.y), i1.y, a1);
      a1 = fmaf(fabsf(xv.z - m1.z), i1.z, a1);
      a1 = fmaf(fabsf(xv.w - m1.w), i1.w, a1);
      accA[r][0] = a0; accA[r][1] = a1;
    }

    // ---- WMMA: M1 (missing mask) @ W, f32 16x16x4 ----
    // A layout: lane l, vgpr j -> (M = l%16, Kdim = j + 2*(l/16))
    v2f av;
    {
      int d0a = d + 2 * hi;                       // d0a&31 <= 30, same word for j=0,1
      unsigned bits = mbits[(rl0 + la) * 8 + (d0a >> 5)];
      av[0] = ((bits >> ( d0a      & 31)) & 1u) ? 1.0f : 0.0f;
      av[1] = ((bits >> ((d0a + 1) & 31)) & 1u) ? 1.0f : 0.0f;
    }
    // B layout: lane l, vgpr j -> (Kdim = j + 2*(l/16), N = l%16)
    #pragma unroll
    for (int t = 0; t < 4; ++t) {
      v2f bv;
      bv[0] = Wm[(d + 2 * hi    ) * K_DIM + t * 16 + la];
      bv[1] = Wm[(d + 2 * hi + 1) * K_DIM + t * 16 + la];
      cacc[t] = __builtin_amdgcn_wmma_f32_16x16x4_f32(
          false, av, false, bv, (short)0, cacc[t], false, false);
    }
  }

  // ---- relayout WMMA C tiles through LDS to lane=k layout ----
  // C layout: vgpr v, lane l -> (M = v + 8*(l/16), N = l%16)
  #pragma unroll
  for (int t = 0; t < 4; ++t) {
    #pragma unroll
    for (int v = 0; v < 8; ++v)
      bsum[wv][(v + 8 * hi) * K_DIM + t * 16 + la] = cacc[t][v];
  }
  __syncthreads();

  const float ck0 = ws[WS_CK + kl0];
  const float ck1 = ws[WS_CK + kl1];

  float wsum = 0.f;
  #pragma unroll
  for (int r = 0; r < ROWS_PER_WAVE; ++r) {
    float e0 = ck0 + bsum[wv][r * K_DIM + kl0] - accA[r][0];
    float e1 = ck1 + bsum[wv][r * K_DIM + kl1] - accA[r][1];
    // logsumexp over 64 values (2 regs x 32 lanes), wave32 butterfly
    float mx = fmaxf(e0, e1);
    for (int off = 16; off > 0; off >>= 1) mx = fmaxf(mx, __shfl_xor(mx, off));
    float s = expf(e0 - mx) + expf(e1 - mx);
    for (int off = 16; off > 0; off >>= 1) s += __shfl_xor(s, off);
    float lse = mx + logf(s);
    float Do = (float)(D_DIM - miss[rl0 + r]);
    wsum += lse - Do * LN2;
  }
  if (lane == 0) partials[blockIdx.x * 2 + wv] = wsum;
}

// ---------------------------------------------------------------------------
// Final: sum partials, subtract N * logsumexp(pi)
__global__ __launch_bounds__(64) void lmm_reduce(const float* __restrict__ pi,
                                                 const float* __restrict__ partials,
                                                 float* __restrict__ out) {
  __shared__ float red[64];
  int tid = threadIdx.x;
  float pv = pi[tid];
  red[tid] = pv; __syncthreads();
  for (int s = 32; s > 0; s >>= 1) {
    if (tid < s) red[tid] = fmaxf(red[tid], red[tid + s]);
    __syncthreads();
  }
  float m = red[0]; __syncthreads();
  red[tid] = expf(pv - m); __syncthreads();
  for (int s = 32; s > 0; s >>= 1) {
    if (tid < s) red[tid] += red[tid + s];
    __syncthreads();
  }
  float lsep = m + logf(red[0]); __syncthreads();

  float s0 = 0.f;
  for (int i = tid; i < 256; i += 64) s0 += partials[i];
  red[tid] = s0; __syncthreads();
  for (int s = 32; s > 0; s >>= 1) {
    if (tid < s) red[tid] += red[tid + s];
    __syncthreads();
  }
  if (tid == 0) out[0] = red[0] - (float)N_ROWS * lsep;
}

// ---------------------------------------------------------------------------
extern "C" void kernel_launch(void* const* d_in, const int* in_sizes, int n_in,
                              void* d_out, int out_size, void* d_ws, size_t ws_size,
                              hipStream_t stream) {
  const float* X  = (const float*)d_in[0];   // [N, D]
  const float* mu = (const float*)d_in[1];   // [D, K]
  const float* b  = (const float*)d_in[2];   // [D, K]
  const float* pi = (const float*)d_in[3];   // [K]
  float* ws  = (float*)d_ws;
  float* out = (float*)d_out;

  lmm_prep<<<(D_DIM * K_DIM) / 256, 256, 0, stream>>>(mu, b, ws);
  lmm_ck<<<1, 64, 0, stream>>>(b, pi, ws);
  lmm_main<<<N_ROWS / ROWS_PER_BLOCK, 64, 0, stream>>>(X, ws, ws + WS_PART);
  lmm_reduce<<<1, 64, 0, stream>>>(pi, ws + WS_PART, out);
}